// QuaternionBatchNorm_8160437862859
// MI455X (gfx1250) — compile-verified
//
#include <hip/hip_runtime.h>

typedef __attribute__((ext_vector_type(2))) float v2f;
typedef __attribute__((ext_vector_type(4))) float v4f;
typedef __attribute__((ext_vector_type(8))) float v8f;

// ---------------------------------------------------------------------------
// Pass 1: per-feature sums (for mean) + raw second moments via f32 WMMA.
// grid = (F/32, NB), block = 256 (8 waves). Each wave owns 4 consecutive
// features = 16 consecutive floats per batch row. K = 4 batch rows per WMMA.
// A(16x4) and B(4x16) register images coincide for this symmetric product:
//   VGPR0 = {lanes0-15: row b0,   lanes16-31: row b0+2} (16 floats each)
//   VGPR1 = {lanes0-15: row b0+1, lanes16-31: row b0+3}
// D(16x16) diagonal 4x4 blocks accumulate the 4 features' x x^T sums.
// ---------------------------------------------------------------------------
__global__ __launch_bounds__(256) void qbn_stats_kernel(
    const float* __restrict__ x, float* __restrict__ ws_cov,
    float* __restrict__ ws_s1, int B, int F, int rows_per_blk)
{
  const int wave = threadIdx.x >> 5;
  const int lane = threadIdx.x & 31;
  const int c    = lane & 15;   // float index within the wave's 16-float group
  const int h    = lane >> 4;   // half-wave select
  const int fbase = blockIdx.x * 32 + wave * 4;
  const long rowf = (long)F * 4;            // floats per batch row
  const long colbase = (long)fbase * 4;     // float offset of this wave's group

  const int bstart = blockIdx.y * rows_per_blk;
  const float* p0 = x + (long)(bstart + 2 * h) * rowf + colbase + c;
  const float* p1 = p0 + rowf;

  v8f  acc = {};     // 16x16 f32 accumulator: block-diag raw second moments
  float s1 = 0.0f;   // per-lane partial sum (for mean)

  const int iters = rows_per_blk >> 2;      // 4 batch rows per WMMA
  for (int it = 0; it < iters; ++it) {
    float a0 = p0[0];
    float a1 = p1[0];
    s1 += a0 + a1;
    v2f a; a[0] = a0; a[1] = a1;
    // D = A * B + C ; A and B are the same registers (symmetric product)
    acc = __builtin_amdgcn_wmma_f32_16x16x4_f32(false, a, false, a,
                                                (short)0, acc, false, false);
    p0 += rowf * 4;
    p1 += rowf * 4;
  }

  // Scatter valid diagonal-block entries. Lane holds D[m,n], m = v+8h, n = c.
  const long base = (long)blockIdx.y * F;
#pragma unroll
  for (int v = 0; v < 8; ++v) {
    int m = v + 8 * h;
    if ((m >> 2) == (c >> 2)) {
      int f = fbase + (m >> 2);
      ws_cov[(base + f) * 16 + (m & 3) * 4 + (c & 3)] = acc[v];
    }
  }
  {
    int f = fbase + (c >> 2);
    ws_s1[((base + f) * 4 + (c & 3)) * 2 + h] = s1;
  }
}

// ---------------------------------------------------------------------------
// Pass 2: one thread per feature. Deterministic reduction of partials,
// cov = M2/B - mu mu^T + eps I, Cholesky, L^-1, fold gamma_sym*L^-1 into a
// single affine transform A x + c with c = beta - A mu. Store [F][20] table.
// ---------------------------------------------------------------------------
__global__ __launch_bounds__(256) void qbn_solve_kernel(
    const float* __restrict__ ws_cov, const float* __restrict__ ws_s1,
    const float* __restrict__ gamma, const float* __restrict__ beta,
    float* __restrict__ ws_A, int B, int F, int NB)
{
  const int f = blockIdx.x * blockDim.x + threadIdx.x;
  if (f >= F) return;
  const float invB = 1.0f / (float)B;

  float s1[4] = {0.f, 0.f, 0.f, 0.f};
  for (int nb = 0; nb < NB; ++nb) {
    const float* p = ws_s1 + ((long)nb * F + f) * 8;
#pragma unroll
    for (int i = 0; i < 4; ++i) s1[i] += p[2 * i] + p[2 * i + 1];
  }
  float mean[4];
#pragma unroll
  for (int i = 0; i < 4; ++i) mean[i] = s1[i] * invB;

  float m2[16] = {};
  for (int nb = 0; nb < NB; ++nb) {
    const float* p = ws_cov + ((long)nb * F + f) * 16;
#pragma unroll
    for (int k = 0; k < 16; ++k) m2[k] += p[k];
  }

  float C[4][4];
#pragma unroll
  for (int i = 0; i < 4; ++i)
#pragma unroll
    for (int j = 0; j < 4; ++j)
      C[i][j] = m2[i * 4 + j] * invB - mean[i] * mean[j];
#pragma unroll
  for (int i = 0; i < 4; ++i) C[i][i] += 1e-5f;

  // Cholesky (lower)
  float L[4][4] = {};
#pragma unroll
  for (int i = 0; i < 4; ++i) {
#pragma unroll
    for (int j = 0; j < 4; ++j) {
      if (j > i) continue;
      float s = C[i][j];
#pragma unroll
      for (int k = 0; k < 4; ++k) if (k < j) s -= L[i][k] * L[j][k];
      L[i][j] = (i == j) ? sqrtf(s) : s / L[j][j];
    }
  }
  // Invert lower triangular via forward substitution
  float Li[4][4] = {};
#pragma unroll
  for (int col = 0; col < 4; ++col) {
#pragma unroll
    for (int i = 0; i < 4; ++i) {
      if (i < col) continue;
      float s = (i == col) ? 1.0f : 0.0f;
#pragma unroll
      for (int k = 0; k < 4; ++k) if (k >= col && k < i) s -= L[i][k] * Li[k][col];
      Li[i][col] = s / L[i][i];
    }
  }
  // Symmetric gamma from tril entries (np.tril_indices(4) order)
  const int tr[10] = {0, 1, 1, 2, 2, 2, 3, 3, 3, 3};
  const int tc[10] = {0, 0, 1, 0, 1, 2, 0, 1, 2, 3};
  float G[4][4];
#pragma unroll
  for (int t = 0; t < 10; ++t) {
    float g = gamma[(long)f * 10 + t];
    G[tr[t]][tc[t]] = g;
    G[tc[t]][tr[t]] = g;
  }
  float A[4][4];
#pragma unroll
  for (int i = 0; i < 4; ++i)
#pragma unroll
    for (int j = 0; j < 4; ++j) {
      float s = 0.f;
#pragma unroll
      for (int k = 0; k < 4; ++k) s += G[i][k] * Li[k][j];
      A[i][j] = s;
    }
  float* pa = ws_A + (long)f * 20;
#pragma unroll
  for (int i = 0; i < 4; ++i) {
    float cv = beta[(long)f * 4 + i];
#pragma unroll
    for (int j = 0; j < 4; ++j) {
      cv -= A[i][j] * mean[j];
      pa[i * 4 + j] = A[i][j];
    }
    pa[16 + i] = cv;
  }
}

// ---------------------------------------------------------------------------
// Pass 3: streaming transform out = A_f x + c_f. grid = (F/32, NB3),
// block = 256. Each thread pins one feature (affine cached in 20 VGPRs) and
// streams b128 load -> 16 FMA -> b128 store; 512B coalesced per wave access.
// ---------------------------------------------------------------------------
__global__ __launch_bounds__(256) void qbn_apply_kernel(
    const float* __restrict__ x, const float* __restrict__ ws_A,
    float* __restrict__ out, int B, int F, int rows_per_blk)
{
  const int f = blockIdx.x * 32 + (threadIdx.x & 31);
  const int rowgrp = threadIdx.x >> 5;  // 0..7

  const float* pa = ws_A + (long)f * 20;
  float A[16], cv[4];
#pragma unroll
  for (int i = 0; i < 16; ++i) A[i] = pa[i];
#pragma unroll
  for (int i = 0; i < 4; ++i) cv[i] = pa[16 + i];

  const v4f* __restrict__ x4 = (const v4f*)x;
  v4f* __restrict__ o4 = (v4f*)out;

  const int bstart = blockIdx.y * rows_per_blk;
  const int iters = rows_per_blk >> 3;  // 8 batch rows per iteration
  long idx = (long)(bstart + rowgrp) * F + f;
  for (int it = 0; it < iters; ++it) {
    if (it + 2 < iters)
      __builtin_prefetch((const void*)(x4 + idx + (long)16 * F), 0, 0);
    v4f v = x4[idx];
    v4f o;
#pragma unroll
    for (int i = 0; i < 4; ++i)
      o[i] = fmaf(A[i * 4 + 0], v[0],
             fmaf(A[i * 4 + 1], v[1],
             fmaf(A[i * 4 + 2], v[2],
             fmaf(A[i * 4 + 3], v[3], cv[i]))));
    o4[idx] = o;
    idx += (long)8 * F;
  }
}

extern "C" void kernel_launch(void* const* d_in, const int* in_sizes, int n_in,
                              void* d_out, int out_size, void* d_ws, size_t ws_size,
                              hipStream_t stream)
{
  const float* x     = (const float*)d_in[0];   // [B, F, 4]
  const float* gamma = (const float*)d_in[1];   // [F, 10]
  const float* beta  = (const float*)d_in[2];   // [F, 4]
  float* out = (float*)d_out;

  const int F = in_sizes[2] / 4;                // 512
  const int B = in_sizes[0] / (F * 4);          // 32768

  const int NB = 64;                            // stats batch blocks
  float* ws_cov = (float*)d_ws;                 // NB*F*16 floats (2 MB)
  float* ws_s1  = ws_cov + (size_t)NB * F * 16; // NB*F*8  floats (1 MB)
  float* ws_A   = ws_s1 + (size_t)NB * F * 8;   // F*20    floats (40 KB)

  dim3 g1(F / 32, NB);
  qbn_stats_kernel<<<g1, 256, 0, stream>>>(x, ws_cov, ws_s1, B, F, B / NB);

  qbn_solve_kernel<<<(F + 255) / 256, 256, 0, stream>>>(
      ws_cov, ws_s1, gamma, beta, ws_A, B, F, NB);

  const int NB3 = 256;                          // apply batch blocks
  dim3 g3(F / 32, NB3);
  qbn_apply_kernel<<<g3, 256, 0, stream>>>(x, ws_A, out, B, F, B / NB3);
}